// AutoFormer_11811160064532
// MI455X (gfx1250) — compile-verified
//
#include <hip/hip_runtime.h>
#include <math.h>

// ---------------------------------------------------------------------------
// Autoformer forward for MI455X (gfx1250, wave32).
// All heavy GEMMs run through V_WMMA_F32_16X16X4_F32 (fp32 matrix pipe).
// FFT autocorrelation is replaced algebraically by a batched Q·K^T GEMM +
// circular-diagonal band reduction (exactly equal to irfft(qf*conj(kf)).mean).
// GEMM: block tile 128x64, 4 waves, wave tile 32x64; A/B tiles staged in LDS
// k-contiguous with stride 18 so every WMMA fragment is one aligned 64-bit
// DS load. Tile fills are register-staged (vector global loads first, DS
// stores after) so loads pipeline instead of serializing on per-element waits.
// ---------------------------------------------------------------------------

typedef __attribute__((ext_vector_type(2))) float v2f;
typedef __attribute__((ext_vector_type(8))) float v8f;

#define NB   16      // batch
#define SL   384     // SEQ == DEC_LEN
#define CIN  21
#define DM   512
#define DFF  2048
#define MARK 4
#define NROW (NB*SL) // 6144
#define TOPK 17      // int(3*ln(384))
#define PREDL 192
#define LABL  192

#define LDSK 18      // padded k-stride (even -> 8B-aligned pairs, good banks)

__device__ __forceinline__ v8f wmma4(v2f a, v2f b, v8f c) {
  // D = A(16x4,f32) * B(4x16,f32) + C(16x16,f32)
  return __builtin_amdgcn_wmma_f32_16x16x4_f32(false, a, false, b, (short)0, c, false, false);
}

__device__ __forceinline__ void store_row16(float* dst, float4 r0, float4 r1,
                                            float4 r2, float4 r3) {
  // dst is 8B-aligned (row stride 72B); emit 8 aligned float2 stores
  float2* d2 = (float2*)dst;
  d2[0] = make_float2(r0.x, r0.y); d2[1] = make_float2(r0.z, r0.w);
  d2[2] = make_float2(r1.x, r1.y); d2[3] = make_float2(r1.z, r1.w);
  d2[4] = make_float2(r2.x, r2.y); d2[5] = make_float2(r2.z, r2.w);
  d2[6] = make_float2(r3.x, r3.y); d2[7] = make_float2(r3.z, r3.w);
}

// ============================ GEMM: C = A*W (+bias)(+gelu)(+resid) ==========
// A: MxK row-major, W: KxN row-major. M%128==0, N%64==0, K%16==0.
// block = 128 threads (4 waves); block tile 128x64; wave computes 32x64.
__global__ void __launch_bounds__(128)
gemm_wmma(const float* __restrict__ A, const float* __restrict__ Bw,
          const float* __restrict__ bias, const float* __restrict__ resid,
          float* __restrict__ C, int M, int N, int K, int gelu)
{
  __shared__ __align__(16) float As[128][LDSK];   // As[m][k]
  __shared__ __align__(16) float Bs[64][LDSK];    // Bs[n][k]  (k-contiguous)
  const int tid  = threadIdx.x;
  const int wave = tid >> 5;
  const int lane = tid & 31;
  const int hl   = lane >> 4;     // lane half
  const int lr   = lane & 15;
  const int rowBase = blockIdx.y * 128;
  const int colBase = blockIdx.x * 64;
  const int mOff = wave * 32;

  v8f acc[2][4];
#pragma unroll
  for (int ms = 0; ms < 2; ms++)
#pragma unroll
    for (int nt = 0; nt < 4; nt++) acc[ms][nt] = v8f{};

  const float* arow = A + (size_t)(rowBase + tid) * K;   // one A row per thread

  for (int k0 = 0; k0 < K; k0 += 16) {
    // ---- load phase (independent vector loads, no interleaved LDS waits) ----
    const float4* ap = (const float4*)(arow + k0);       // 16B aligned
    float4 r0 = ap[0], r1 = ap[1], r2 = ap[2], r3 = ap[3];
    float breg[8];
#pragma unroll
    for (int i = 0; i < 8; i++) {                        // B 16x64, coalesced in n
      int e = tid + i * 128;
      breg[i] = Bw[(size_t)(k0 + (e >> 6)) * N + colBase + (e & 63)];
    }
    if (k0 + 16 < K) {                                   // prefetch next tiles
      __builtin_prefetch(&arow[k0 + 16], 0, 3);
      __builtin_prefetch(&Bw[(size_t)(k0 + 16 + (tid & 15)) * N + colBase], 0, 3);
    }
    // ---- store phase ----
    store_row16(&As[tid][0], r0, r1, r2, r3);
#pragma unroll
    for (int i = 0; i < 8; i++) {
      int e = tid + i * 128;
      Bs[e & 63][e >> 6] = breg[i];
    }
    __syncthreads();
#pragma unroll
    for (int ks = 0; ks < 16; ks += 4) {
      v2f a0, a1;
      a0.x = As[mOff + lr     ][ks + 2 * hl];
      a0.y = As[mOff + lr     ][ks + 2 * hl + 1];
      a1.x = As[mOff + 16 + lr][ks + 2 * hl];
      a1.y = As[mOff + 16 + lr][ks + 2 * hl + 1];
#pragma unroll
      for (int nt = 0; nt < 4; nt++) {
        v2f b;
        b.x = Bs[nt * 16 + lr][ks + 2 * hl];
        b.y = Bs[nt * 16 + lr][ks + 2 * hl + 1];
        acc[0][nt] = wmma4(a0, b, acc[0][nt]);
        acc[1][nt] = wmma4(a1, b, acc[1][nt]);
      }
    }
    __syncthreads();
  }
#pragma unroll
  for (int ms = 0; ms < 2; ms++) {
#pragma unroll
    for (int nt = 0; nt < 4; nt++) {
      int col = colBase + nt * 16 + lr;
      float bv = bias ? bias[col] : 0.0f;
#pragma unroll
      for (int i = 0; i < 8; i++) {
        int row = rowBase + mOff + ms * 16 + hl * 8 + i;
        float v = acc[ms][nt][i] + bv;
        if (gelu) v = 0.5f * v * (1.0f + erff(v * 0.70710678118654752f));
        if (resid) v += resid[(size_t)row * N + col];
        C[(size_t)row * N + col] = v;
      }
    }
  }
}

// ================= batched R[b] = Q[b] * K[b]^T  (384x384, K=512) ===========
__global__ void __launch_bounds__(128)
corr_gemm_wmma(const float* __restrict__ Q, const float* __restrict__ Km,
               float* __restrict__ R)
{
  __shared__ __align__(16) float As[128][LDSK];
  __shared__ __align__(16) float Bs[64][LDSK];
  const int tid  = threadIdx.x;
  const int wave = tid >> 5;
  const int lane = tid & 31;
  const int hl   = lane >> 4;
  const int lr   = lane & 15;
  const int b    = blockIdx.z;
  const float* Ab = Q  + (size_t)b * SL * DM;
  const float* Bb = Km + (size_t)b * SL * DM;
  float*       Cb = R  + (size_t)b * SL * SL;
  const int rowBase = blockIdx.y * 128;
  const int colBase = blockIdx.x * 64;
  const int mOff = wave * 32;

  v8f acc[2][4];
#pragma unroll
  for (int ms = 0; ms < 2; ms++)
#pragma unroll
    for (int nt = 0; nt < 4; nt++) acc[ms][nt] = v8f{};

  const float* arow = Ab + (size_t)(rowBase + tid) * DM;       // one A row per thread
  const int    bn   = tid >> 1;                                // half B row per thread
  const int    bkh  = (tid & 1) * 8;
  const float* brow = Bb + (size_t)(colBase + bn) * DM + bkh;

  for (int k0 = 0; k0 < DM; k0 += 16) {
    const float4* ap = (const float4*)(arow + k0);
    float4 r0 = ap[0], r1 = ap[1], r2 = ap[2], r3 = ap[3];
    const float4* bp = (const float4*)(brow + k0);             // B already n-major
    float4 s0 = bp[0], s1 = bp[1];

    store_row16(&As[tid][0], r0, r1, r2, r3);
    float2* bd = (float2*)&Bs[bn][bkh];                        // 8B aligned
    bd[0] = make_float2(s0.x, s0.y); bd[1] = make_float2(s0.z, s0.w);
    bd[2] = make_float2(s1.x, s1.y); bd[3] = make_float2(s1.z, s1.w);
    __syncthreads();
#pragma unroll
    for (int ks = 0; ks < 16; ks += 4) {
      v2f a0, a1;
      a0.x = As[mOff + lr     ][ks + 2 * hl];
      a0.y = As[mOff + lr     ][ks + 2 * hl + 1];
      a1.x = As[mOff + 16 + lr][ks + 2 * hl];
      a1.y = As[mOff + 16 + lr][ks + 2 * hl + 1];
#pragma unroll
      for (int nt = 0; nt < 4; nt++) {
        v2f bb;
        bb.x = Bs[nt * 16 + lr][ks + 2 * hl];
        bb.y = Bs[nt * 16 + lr][ks + 2 * hl + 1];
        acc[0][nt] = wmma4(a0, bb, acc[0][nt]);
        acc[1][nt] = wmma4(a1, bb, acc[1][nt]);
      }
    }
    __syncthreads();
  }
#pragma unroll
  for (int ms = 0; ms < 2; ms++) {
#pragma unroll
    for (int nt = 0; nt < 4; nt++) {
      int col = colBase + nt * 16 + lr;
#pragma unroll
      for (int i = 0; i < 8; i++) {
        int row = rowBase + mOff + ms * 16 + hl * 8 + i;
        Cb[(size_t)row * SL + col] = acc[ms][nt][i];
      }
    }
  }
}

// mean_value[b,tau] = (1/512) * sum_t R[b, (t+tau)%L, t]
__global__ void band_mean(const float* __restrict__ R, float* __restrict__ mv)
{
  int i = blockIdx.x * blockDim.x + threadIdx.x;
  if (i >= NB * SL) return;
  int b = i / SL, tau = i % SL;
  const float* Rb = R + (size_t)b * SL * SL;
  float s = 0.0f;
  for (int t = 0; t < SL; t++) {
    int r = t + tau; if (r >= SL) r -= SL;
    s += Rb[(size_t)r * SL + t];
  }
  mv[i] = s * (1.0f / 512.0f);
}

// top-17 indices of sum_b mv[b,tau]
__global__ void __launch_bounds__(512)
topk17(const float* __restrict__ mv, int* __restrict__ idx)
{
  __shared__ float g[512];
  __shared__ float rv[512];
  __shared__ int   ri[512];
  int t = threadIdx.x;
  float s = 0.0f;
  if (t < SL) for (int b = 0; b < NB; b++) s += mv[b * SL + t];
  g[t] = (t < SL) ? s : -1e30f;
  __syncthreads();
  for (int k = 0; k < TOPK; k++) {
    rv[t] = g[t]; ri[t] = t;
    __syncthreads();
    for (int off = 256; off > 0; off >>= 1) {
      if (t < off && rv[t + off] > rv[t]) { rv[t] = rv[t + off]; ri[t] = ri[t + off]; }
      __syncthreads();
    }
    if (t == 0) { idx[k] = ri[0]; g[ri[0]] = -1e30f; }
    __syncthreads();
  }
}

// w[b,k] = softmax_k( mv[b, idx[k]] )
__global__ void softmax17(const float* __restrict__ mv, const int* __restrict__ idx,
                          float* __restrict__ w)
{
  int b = threadIdx.x;
  if (b >= NB) return;
  float vals[TOPK], m = -1e30f;
  for (int k = 0; k < TOPK; k++) { vals[k] = mv[b * SL + idx[k]]; m = fmaxf(m, vals[k]); }
  float s = 0.0f;
  for (int k = 0; k < TOPK; k++) { vals[k] = expf(vals[k] - m); s += vals[k]; }
  for (int k = 0; k < TOPK; k++) w[b * 32 + k] = vals[k] / s;
}

// out[b,l,:] = sum_k w[b,k] * V[b,(l+idx_k)%L,:]
__global__ void __launch_bounds__(256)
agg_roll(const float* __restrict__ V, const float* __restrict__ w,
         const int* __restrict__ idx, float* __restrict__ out)
{
  int row = blockIdx.x;            // b*SL + l
  int b = row / SL, l = row % SL;
  __shared__ float ws[TOPK];
  __shared__ int   rs[TOPK];
  if (threadIdx.x < TOPK) {
    ws[threadIdx.x] = w[b * 32 + threadIdx.x];
    int r = l + idx[threadIdx.x]; if (r >= SL) r -= SL;
    rs[threadIdx.x] = r;
  }
  __syncthreads();
  for (int d = threadIdx.x; d < DM; d += 256) {
    float s = 0.0f;
#pragma unroll
    for (int k = 0; k < TOPK; k++)
      s += ws[k] * V[((size_t)b * SL + rs[k]) * DM + d];
    out[(size_t)row * DM + d] = s;
  }
}

// series_decomp: moving average window 25, replicate padding.
// trend_mode: 0=discard trend, 1=store, 2=accumulate
__global__ void decomp_k(const float* __restrict__ x, float* __restrict__ seas,
                         float* __restrict__ trend, int Cc, int trend_mode)
{
  int i = blockIdx.x * blockDim.x + threadIdx.x;
  int total = NB * SL * Cc;
  if (i >= total) return;
  int c = i % Cc; int rest = i / Cc; int l = rest % SL; int b = rest / SL;
  float s = 0.0f;
  for (int j = -12; j <= 12; j++) {
    int ll = l + j; ll = ll < 0 ? 0 : (ll >= SL ? SL - 1 : ll);
    s += x[((size_t)b * SL + ll) * Cc + c];
  }
  float mean = s * (1.0f / 25.0f);
  seas[i] = x[i] - mean;
  if (trend_mode == 1) trend[i] = mean;
  else if (trend_mode == 2) trend[i] += mean;
}

// layernorm over last dim (D=512), one block per row
__global__ void __launch_bounds__(256)
ln_row(const float* __restrict__ x, const float* __restrict__ w,
       const float* __restrict__ bb, float* __restrict__ out)
{
  __shared__ float red[256];
  int row = blockIdx.x;
  const float* xr = x + (size_t)row * DM;
  float s = 0.0f;
  for (int d = threadIdx.x; d < DM; d += 256) s += xr[d];
  red[threadIdx.x] = s; __syncthreads();
  for (int off = 128; off > 0; off >>= 1) {
    if (threadIdx.x < off) red[threadIdx.x] += red[threadIdx.x + off];
    __syncthreads();
  }
  float mu = red[0] * (1.0f / DM);
  __syncthreads();
  float s2 = 0.0f;
  for (int d = threadIdx.x; d < DM; d += 256) { float t = xr[d] - mu; s2 += t * t; }
  red[threadIdx.x] = s2; __syncthreads();
  for (int off = 128; off > 0; off >>= 1) {
    if (threadIdx.x < off) red[threadIdx.x] += red[threadIdx.x + off];
    __syncthreads();
  }
  float inv = rsqrtf(red[0] * (1.0f / DM) + 1e-5f);
  for (int d = threadIdx.x; d < DM; d += 256)
    out[(size_t)row * DM + d] = (xr[d] - mu) * inv * w[d] + bb[d];
}

// subtract per-(b,d) mean over sequence axis (my_layernorm's second stage)
__global__ void col_center(float* __restrict__ x)
{
  int i = blockIdx.x * blockDim.x + threadIdx.x;
  if (i >= NB * DM) return;
  int b = i / DM, d = i % DM;
  float s = 0.0f;
  for (int l = 0; l < SL; l++) s += x[((size_t)b * SL + l) * DM + d];
  float m = s * (1.0f / SL);
  for (int l = 0; l < SL; l++) x[((size_t)b * SL + l) * DM + d] -= m;
}

// embed = circ_conv3(x, Wtok) + mark @ Wtemp
__global__ void embed_k(const float* __restrict__ x, const float* __restrict__ mark,
                        const float* __restrict__ Wtok, const float* __restrict__ Wtemp,
                        float* __restrict__ out)
{
  int i = blockIdx.x * blockDim.x + threadIdx.x;
  if (i >= NB * SL * DM) return;
  int d = i % DM; int row = i / DM; int l = row % SL; int b = row / SL;
  int lm = (l == 0) ? SL - 1 : l - 1;
  int lp = (l == SL - 1) ? 0 : l + 1;
  const float* x0 = x + ((size_t)b * SL + lm) * CIN;
  const float* x1 = x + ((size_t)b * SL + l ) * CIN;
  const float* x2 = x + ((size_t)b * SL + lp) * CIN;
  float s = 0.0f;
#pragma unroll
  for (int c = 0; c < CIN; c++) {
    s += x0[c] * Wtok[(0 * CIN + c) * DM + d];
    s += x1[c] * Wtok[(1 * CIN + c) * DM + d];
    s += x2[c] * Wtok[(2 * CIN + c) * DM + d];
  }
  const float* mk = mark + ((size_t)b * SL + l) * MARK;
#pragma unroll
  for (int m = 0; m < MARK; m++) s += mk[m] * Wtemp[m * DM + d];
  out[i] = s;
}

// trend projection: tacc += circ_conv3(T, Wtrend)   (D -> CIN)
__global__ void trend_conv(const float* __restrict__ T, const float* __restrict__ Wt,
                           float* __restrict__ tacc)
{
  int i = blockIdx.x * blockDim.x + threadIdx.x;
  if (i >= NB * SL * CIN) return;
  int c = i % CIN; int row = i / CIN; int l = row % SL; int b = row / SL;
  int lm = (l == 0) ? SL - 1 : l - 1;
  int lp = (l == SL - 1) ? 0 : l + 1;
  const float* t0 = T + ((size_t)b * SL + lm) * DM;
  const float* t1 = T + ((size_t)b * SL + l ) * DM;
  const float* t2 = T + ((size_t)b * SL + lp) * DM;
  float s = 0.0f;
  for (int d = 0; d < DM; d++) {
    s += t0[d] * Wt[(0 * DM + d) * CIN + c];
    s += t1[d] * Wt[(1 * DM + d) * CIN + c];
    s += t2[d] * Wt[(2 * DM + d) * CIN + c];
  }
  tacc[i] += s;
}

// per-(b,c) mean over sequence of x_enc
__global__ void meanx_k(const float* __restrict__ x, float* __restrict__ mx)
{
  int i = blockIdx.x * blockDim.x + threadIdx.x;
  if (i >= NB * CIN) return;
  int b = i / CIN, c = i % CIN;
  float s = 0.0f;
  for (int l = 0; l < SL; l++) s += x[((size_t)b * SL + l) * CIN + c];
  mx[i] = s * (1.0f / SL);
}

// seasonal_init and trend_init(=initial trend_acc)
__global__ void build_inits(const float* __restrict__ seas, const float* __restrict__ trend,
                            const float* __restrict__ mx, float* __restrict__ sinit,
                            float* __restrict__ tacc)
{
  int i = blockIdx.x * blockDim.x + threadIdx.x;
  if (i >= NB * SL * CIN) return;
  int c = i % CIN; int row = i / CIN; int l = row % SL; int b = row / SL;
  if (l < LABL) {
    sinit[i] = seas [((size_t)b * SL + (SL - LABL + l)) * CIN + c];
    tacc [i] = trend[((size_t)b * SL + (SL - LABL + l)) * CIN + c];
  } else {
    sinit[i] = 0.0f;
    tacc [i] = mx[b * CIN + c];
  }
}

// out = (trend_acc + dec @ proj_W + proj_b)[:, -PRED:]
__global__ void proj_out(const float* __restrict__ dec, const float* __restrict__ Wp,
                         const float* __restrict__ bp, const float* __restrict__ tacc,
                         float* __restrict__ out)
{
  int i = blockIdx.x * blockDim.x + threadIdx.x;
  if (i >= NB * PREDL * CIN) return;
  int c = i % CIN; int r = i / CIN; int lp = r % PREDL; int b = r / PREDL;
  int l = LABL + lp;
  const float* xr = dec + ((size_t)b * SL + l) * DM;
  float s = bp[c];
  for (int d = 0; d < DM; d++) s += xr[d] * Wp[d * CIN + c];
  out[i] = s + tacc[((size_t)b * SL + l) * CIN + c];
}

// ============================ host orchestration ============================
struct AttnP { const float *Wq,*bq,*Wk,*bk,*Wv,*bv,*Wo,*bo; };

static void run_attn(const float* xq, const float* xkv, const AttnP& P,
                     float* Q, float* Kb, float* Vb, float* R, float* MV,
                     float* W17, int* IDX, float* AGG, float* OUT, hipStream_t s)
{
  dim3 gproj(DM / 64, NROW / 128);
  gemm_wmma<<<gproj, 128, 0, s>>>(xq,  P.Wq, P.bq, nullptr, Q,  NROW, DM, DM, 0);
  gemm_wmma<<<gproj, 128, 0, s>>>(xkv, P.Wk, P.bk, nullptr, Kb, NROW, DM, DM, 0);
  gemm_wmma<<<gproj, 128, 0, s>>>(xkv, P.Wv, P.bv, nullptr, Vb, NROW, DM, DM, 0);
  corr_gemm_wmma<<<dim3(SL / 64, SL / 128, NB), 128, 0, s>>>(Q, Kb, R);
  band_mean<<<(NB * SL + 255) / 256, 256, 0, s>>>(R, MV);
  topk17<<<1, 512, 0, s>>>(MV, IDX);
  softmax17<<<1, 32, 0, s>>>(MV, IDX, W17);
  agg_roll<<<NROW, 256, 0, s>>>(Vb, W17, IDX, AGG);
  gemm_wmma<<<gproj, 128, 0, s>>>(AGG, P.Wo, P.bo, xq, OUT, NROW, DM, DM, 0);
}

extern "C" void kernel_launch(void* const* d_in, const int* in_sizes, int n_in,
                              void* d_out, int out_size, void* d_ws, size_t ws_size,
                              hipStream_t stream)
{
  (void)in_sizes; (void)n_in; (void)out_size; (void)ws_size;
  // ---- inputs (jax pytree order: dicts flattened by sorted key) ----
  const float* x_enc  = (const float*)d_in[0];
  const float* xm_enc = (const float*)d_in[1];
  /* d_in[2] = x_dec is unused by the reference */
  const float* xm_dec = (const float*)d_in[3];
  int p = 4;
  auto IN = [&](void) { return (const float*)d_in[p++]; };
  const float* dec_Wtemp = IN(); const float* dec_Wtok = IN();
  const float* dW1 = IN(); const float* dW2 = IN(); const float* dWtrend = IN();
  AttnP dcross; dcross.Wk = IN(); dcross.Wo = IN(); dcross.Wq = IN(); dcross.Wv = IN();
                dcross.bk = IN(); dcross.bo = IN(); dcross.bq = IN(); dcross.bv = IN();
  AttnP dself;  dself.Wk  = IN(); dself.Wo  = IN(); dself.Wq  = IN(); dself.Wv  = IN();
                dself.bk  = IN(); dself.bo  = IN(); dself.bq  = IN(); dself.bv  = IN();
  const float* dec_nb = IN(); const float* dec_nw = IN();
  const float* enc_Wtemp = IN(); const float* enc_Wtok = IN();
  AttnP eatt[2]; const float *eW1[2], *eW2[2];
  for (int li = 0; li < 2; li++) {
    eW1[li] = IN(); eW2[li] = IN();
    eatt[li].Wk = IN(); eatt[li].Wo = IN(); eatt[li].Wq = IN(); eatt[li].Wv = IN();
    eatt[li].bk = IN(); eatt[li].bo = IN(); eatt[li].bq = IN(); eatt[li].bv = IN();
  }
  const float* enc_nb = IN(); const float* enc_nw = IN();
  const float* proj_W = IN(); const float* proj_b = IN();

  // ---- workspace arena (~150 MB of f32) ----
  float* W = (float*)d_ws;
  size_t off = 0;
  auto alloc = [&](size_t n) { float* q = W + off; off += n; return q; };
  float* A0   = alloc((size_t)NROW * DM);
  float* A1   = alloc((size_t)NROW * DM);
  float* Qb   = alloc((size_t)NROW * DM);
  float* Kb   = alloc((size_t)NROW * DM);
  float* Vb   = alloc((size_t)NROW * DM);
  float* ENC  = alloc((size_t)NROW * DM);
  float* Tsum = alloc((size_t)NROW * DM);
  float* F    = alloc((size_t)NROW * DFF);
  float* R    = alloc((size_t)NB * SL * SL);
  float* MV   = alloc((size_t)NB * SL);
  float* W17  = alloc((size_t)NB * 32);
  float* SEAS = alloc((size_t)NB * SL * CIN);
  float* TRND = alloc((size_t)NB * SL * CIN);
  float* SINIT= alloc((size_t)NB * SL * CIN);
  float* TACC = alloc((size_t)NB * SL * CIN);
  float* MX   = alloc((size_t)NB * CIN);
  int*   IDX  = (int*)alloc(32);

  const int TPB = 256;
  const int gBLD = (NB * SL * DM  + TPB - 1) / TPB;   // B*L*D grid
  const int gBLC = (NB * SL * CIN + TPB - 1) / TPB;   // B*L*CIN grid
  dim3 gffn1(DFF / 64, NROW / 128);
  dim3 gffn2(DM  / 64, NROW / 128);

  // ======================= ENCODER =======================
  embed_k<<<gBLD, TPB, 0, stream>>>(x_enc, xm_enc, enc_Wtok, enc_Wtemp, A0);
  for (int li = 0; li < 2; li++) {
    run_attn(A0, A0, eatt[li], Qb, Kb, Vb, R, MV, W17, IDX, A1, Qb, stream); // Qb = x+attn
    decomp_k<<<gBLD, TPB, 0, stream>>>(Qb, A0, nullptr, DM, 0);              // x = seasonal
    gemm_wmma<<<gffn1, 128, 0, stream>>>(A0, eW1[li], nullptr, nullptr, F, NROW, DFF, DM, 1);
    gemm_wmma<<<gffn2, 128, 0, stream>>>(F, eW2[li], nullptr, A0, A1, NROW, DM, DFF, 0);
    decomp_k<<<gBLD, TPB, 0, stream>>>(A1, A0, nullptr, DM, 0);
  }
  ln_row<<<NROW, 256, 0, stream>>>(A0, enc_nw, enc_nb, ENC);
  col_center<<<(NB * DM + TPB - 1) / TPB, TPB, 0, stream>>>(ENC);

  // ======================= DECODER INIT =======================
  decomp_k<<<gBLC, TPB, 0, stream>>>(x_enc, SEAS, TRND, CIN, 1);
  meanx_k<<<(NB * CIN + TPB - 1) / TPB, TPB, 0, stream>>>(x_enc, MX);
  build_inits<<<gBLC, TPB, 0, stream>>>(SEAS, TRND, MX, SINIT, TACC);
  embed_k<<<gBLD, TPB, 0, stream>>>(SINIT, xm_dec, dec_Wtok, dec_Wtemp, A0);

  // ======================= DECODER LAYER =======================
  run_attn(A0, A0, dself, Qb, Kb, Vb, R, MV, W17, IDX, A1, Qb, stream);    // x + self_attn
  decomp_k<<<gBLD, TPB, 0, stream>>>(Qb, A0, Tsum, DM, 1);                 // t1 -> Tsum
  run_attn(A0, ENC, dcross, Qb, Kb, Vb, R, MV, W17, IDX, A1, Qb, stream);  // x + cross_attn
  decomp_k<<<gBLD, TPB, 0, stream>>>(Qb, A0, Tsum, DM, 2);                 // Tsum += t2
  gemm_wmma<<<gffn1, 128, 0, stream>>>(A0, dW1, nullptr, nullptr, F, NROW, DFF, DM, 1);
  gemm_wmma<<<gffn2, 128, 0, stream>>>(F, dW2, nullptr, A0, A1, NROW, DM, DFF, 0);
  decomp_k<<<gBLD, TPB, 0, stream>>>(A1, A0, Tsum, DM, 2);                 // Tsum += t3
  trend_conv<<<gBLC, TPB, 0, stream>>>(Tsum, dWtrend, TACC);               // TACC += rt

  // ======================= OUTPUT =======================
  ln_row<<<NROW, 256, 0, stream>>>(A0, dec_nw, dec_nb, A1);
  col_center<<<(NB * DM + TPB - 1) / TPB, TPB, 0, stream>>>(A1);
  proj_out<<<(NB * PREDL * CIN + TPB - 1) / TPB, TPB, 0, stream>>>(
      A1, proj_W, proj_b, TACC, (float*)d_out);
}